// QueryAndGroupDilated_31576599560762
// MI455X (gfx1250) — compile-verified
//
#include <hip/hip_runtime.h>

// Problem constants (from reference setup_inputs)
#define B_ 2
#define N_ 16384
#define M_ 2048
#define C_ 64
#define S_ 32         // NSAMPLE
#define SEGS_ 8       // waves per block = N segments per query tile
#define SEGN_ (N_ / SEGS_)
#define CHUNKS_ (SEGN_ / 16)

typedef __attribute__((ext_vector_type(2))) float v2f;
typedef __attribute__((ext_vector_type(8))) float v8f;

__device__ __forceinline__ unsigned ballot32(bool p) {
#if __has_builtin(__builtin_amdgcn_ballot_w32)
    return __builtin_amdgcn_ballot_w32(p);
#else
    return (unsigned)__ballot(p ? 1 : 0);
#endif
}

// ---------------------------------------------------------------------------
// Kernel 1: dilated ball query via V_WMMA_F32_16X16X4_F32.
// One block per 16-query tile; 8 waves each scan a contiguous N/8 segment
// (ordering preserved: all matches of segment i precede those of segment i+1).
//   A (16x4 f32): lane<16 -> (q.x, q.y), lane>=16 -> (q.z, 0)  [K padded to 4]
//   B (4x16 f32): lane<16 -> (p.x, p.y), lane>=16 -> (p.z, 0)
//   D (16x16 f32): acc[r] = dot(q[row], p[col]), row = r + 8*(lane>=16),
//                  col = lane & 15
// d2 = |q|^2 + |p|^2 - 2*dot. Per-row running counts are wave-uniform
// (derived from the uniform ballot) -> kept in SGPRs, no shuffles needed.
// A block-level ordered prefix-merge then produces the final idx/cnt.
// ---------------------------------------------------------------------------
__global__ __launch_bounds__(256) void bq_wmma_kernel(
        const float* __restrict__ xyz,      // (B,N,3)
        const float* __restrict__ new_xyz,  // (B,M,3)
        float* __restrict__ out_cnt,        // (B,M) -> d_out head
        int*   __restrict__ ws_idx)         // (B,M,S) scratch
{
    __shared__ float s_q[16][4];            // queries: x,y,z,|q|^2
    __shared__ int   s_seg[SEGS_][16][S_];  // per-segment first <=32 matches
    __shared__ int   s_cnt[SEGS_][16];      // per-segment uncapped counts

    const int t    = threadIdx.x;
    const int lane = t & 31;
    const int wv   = t >> 5;                // segment id
    const int tile = blockIdx.x;            // 0 .. B*M/16-1
    const int b    = tile / (M_ / 16);
    const int m0   = (tile % (M_ / 16)) * 16;

    const float* __restrict__ xb = xyz     + (size_t)b * N_ * 3;
    const float* __restrict__ qb = new_xyz + (size_t)b * M_ * 3;

    if (t < 16) {
        const float qx = qb[(m0 + t) * 3 + 0];
        const float qy = qb[(m0 + t) * 3 + 1];
        const float qz = qb[(m0 + t) * 3 + 2];
        s_q[t][0] = qx; s_q[t][1] = qy; s_q[t][2] = qz;
        s_q[t][3] = qx*qx + qy*qy + qz*qz;
    }
    __syncthreads();

    const int  col = lane & 15;
    const bool hi  = (lane >= 16);

    v2f A;
    {
        const float qx = s_q[col][0];
        const float qy = s_q[col][1];
        const float qz = s_q[col][2];
        A.x = hi ? qz : qx;
        A.y = hi ? 0.0f : qy;
    }
    float qn2[8];
#pragma unroll
    for (int r = 0; r < 8; ++r) qn2[r] = s_q[r + (hi ? 8 : 0)][3];

    const float rin2  = 0.05f * 0.05f;
    const float rout2 = 0.20f * 0.20f;
    const unsigned halfsel = hi ? 0xFFFF0000u : 0x0000FFFFu;
    const unsigned ltmask  = (1u << lane) - 1u;   // bits [0, lane)
    const unsigned lanebit = 1u << lane;

    // wave-uniform running counts: rows r (cLo) and r+8 (cHi) -> SGPRs
    int cLo[8], cHi[8];
#pragma unroll
    for (int r = 0; r < 8; ++r) { cLo[r] = 0; cHi[r] = 0; }

    const float* pcur = xb + (size_t)(wv * SEGN_ + col) * 3;
    int nidx = wv * SEGN_ + col;            // this lane's point index

    for (int it = 0; it < CHUNKS_; ++it) {  // constant trip count -> SALU loop
        // speculative prefetch of next chunk; translation faults are dropped,
        // so no guard (and no EXEC manipulation) is needed.
        __builtin_prefetch(pcur + 16 * 3, 0, 1);   // global_prefetch_b8

        const float px = pcur[0];
        const float py = pcur[1];
        const float pz = pcur[2];
        v2f Bv;
        Bv.x = hi ? pz : px;
        Bv.y = hi ? 0.0f : py;
        const float pn2 = px*px + py*py + pz*pz;

        v8f acc = {0.f, 0.f, 0.f, 0.f, 0.f, 0.f, 0.f, 0.f};
        acc = __builtin_amdgcn_wmma_f32_16x16x4_f32(
                  /*neg_a=*/false, A, /*neg_b=*/false, Bv,
                  /*c_mod=*/(short)0, acc, /*reuse_a=*/false, /*reuse_b=*/false);

#pragma unroll
        for (int r = 0; r < 8; ++r) {
            const float d2 = qn2[r] + pn2 - 2.0f * acc[r];
            // feed the compare straight into ballot: no bool materialization
            const unsigned bal = ballot32((d2 >= rin2) && (d2 < rout2));
            if (bal) {  // uniform branch; ~40% taken at ~3% hit rate
                const int base = hi ? cHi[r] : cLo[r];   // cndmask of SGPRs
                const int slot = base + __popc(bal & halfsel & ltmask);
                if ((bal & lanebit) && slot < S_)
                    s_seg[wv][r + (hi ? 8 : 0)][slot] = nidx;
                cLo[r] += __popc(bal & 0xFFFFu);
                cHi[r] += __popc(bal >> 16);
            }
        }
        pcur += 16 * 3;
        nidx += 16;
    }

    // publish per-row segment counts (uniform scalars -> per-lane select chain)
    int myc = 0;
#pragma unroll
    for (int r = 0; r < 8; ++r) {
        myc = (lane == r)     ? cLo[r] : myc;
        myc = (lane == r + 8) ? cHi[r] : myc;
    }
    if (lane < 16) s_cnt[wv][lane] = myc;
    __syncthreads();

    // ordered prefix-merge of the 8 segments: 16 rows x 32 slots = 512 pairs
    const int gmb = b * M_ + m0;
    for (int pair = t; pair < 16 * S_; pair += 256) {
        const int row  = pair >> 5;
        const int slot = pair & 31;
        int c[SEGS_], total = 0;
#pragma unroll
        for (int s = 0; s < SEGS_; ++s) { c[s] = s_cnt[s][row]; total += c[s]; }
        const int cnt_out = total < S_ ? total : S_;
        int e = 0;
        if (slot < cnt_out) {
            int acc = 0;
#pragma unroll
            for (int s = 0; s < SEGS_; ++s) {
                if (slot >= acc && slot < acc + c[s])
                    e = s_seg[s][row][slot - acc];
                acc += c[s];
            }
        } else if (total > 0) {   // back-fill with first matched index
            bool done = false;
#pragma unroll
            for (int s = 0; s < SEGS_; ++s) {
                if (!done && c[s] > 0) { e = s_seg[s][row][0]; done = true; }
            }
        }
        ws_idx[(gmb + row) * S_ + slot] = e;
        if (slot == 0) out_cnt[gmb + row] = (float)cnt_out;
    }
}

// ---------------------------------------------------------------------------
// Kernel 2: gather + center + concat. HBM-write-bound (~35 MB out); all
// gather sources fit in the 192 MB L2. Lane = slot s -> 128B fully coalesced
// stores per (b,c,m) row. 512 blocks x 256 threads.
// Output layout: d_out = [ (B,M) counts | (B, 3+C, M, S) features ].
// ---------------------------------------------------------------------------
__global__ __launch_bounds__(256) void group_gather_kernel(
        const float* __restrict__ xyz,      // (B,N,3)
        const float* __restrict__ new_xyz,  // (B,M,3)
        const float* __restrict__ feat,     // (B,C,N)
        const int*   __restrict__ ws_idx,   // (B,M,S)
        float* __restrict__ out)
{
    const int t  = threadIdx.x;
    const int s  = t & 31;
    const int ml = t >> 5;                 // 0..7
    const int gm = blockIdx.x * 8 + ml;    // 0 .. B*M-1
    const int b  = gm / M_;
    const int m  = gm - b * M_;
    const int j  = ws_idx[gm * S_ + s];

    float* __restrict__ of = out + (size_t)B_ * M_;
    const float* __restrict__ xb = xyz + (size_t)b * N_ * 3;

    const float qx = new_xyz[(size_t)gm * 3 + 0];
    const float qy = new_xyz[(size_t)gm * 3 + 1];
    const float qz = new_xyz[(size_t)gm * 3 + 2];
    const float px = xb[(size_t)j * 3 + 0];
    const float py = xb[(size_t)j * 3 + 1];
    const float pz = xb[(size_t)j * 3 + 2];

    const size_t chan = (size_t)M_ * S_;
    size_t o = ((size_t)(b * (C_ + 3)) * M_ + m) * S_ + s;
    of[o] = px - qx; o += chan;
    of[o] = py - qy; o += chan;
    of[o] = pz - qz; o += chan;

    const float* __restrict__ fb = feat + (size_t)b * C_ * N_ + j;
#pragma unroll 8
    for (int c = 0; c < C_; ++c) {
        of[o] = fb[(size_t)c * N_];
        o += chan;
    }
}

extern "C" void kernel_launch(void* const* d_in, const int* in_sizes, int n_in,
                              void* d_out, int out_size, void* d_ws, size_t ws_size,
                              hipStream_t stream) {
    const float* xyz     = (const float*)d_in[0];
    const float* new_xyz = (const float*)d_in[1];
    const float* feat    = (const float*)d_in[2];
    float* out   = (float*)d_out;
    int*   idxws = (int*)d_ws;   // needs B*M*S*4 = 512 KB

    // one block per 16-query tile -> 256 blocks x 8 waves (full WGP coverage)
    bq_wmma_kernel<<<dim3(B_ * (M_ / 16)), dim3(256), 0, stream>>>(
        xyz, new_xyz, out, idxws);
    // 8 queries/block * 32 slots -> 512 blocks
    group_gather_kernel<<<dim3((B_ * M_) / 8), dim3(256), 0, stream>>>(
        xyz, new_xyz, feat, idxws, out);
}